// VideoHungarianMatcherProjMask_85650237817561
// MI455X (gfx1250) — compile-verified
//
#include <hip/hip_runtime.h>
#include <hip/hip_bf16.h>

// ---------------- problem constants ----------------
constexpr int B = 2, Q = 100, NC = 41, G = 12, T = 6, H = 96, W = 160;
constexpr int HW  = H * W;
constexpr int KW_ = T * H;   // 576  (K for W-projection numerators)
constexpr int KH_ = T * W;   // 960  (K for H-projection numerators)
constexpr int QPAD = 112, GPAD = 16, NTERM = 4;

// ---------------- workspace layout (float offsets) ----------------
constexpr size_t OFF_S    = 0;                                   // [B][Q][G][4] pairwise masked-proj sums
constexpr size_t OFF_TS   = OFF_S    + (size_t)B * Q * G * 4;    // [B][G][4]   tgt-proj sums
constexpr size_t OFF_ZPAD = OFF_TS   + (size_t)B * G * 4;        // [KH_] zero row for padded-g WMMA lanes
constexpr size_t OFF_NUM  = OFF_ZPAD + (size_t)KH_;              // [B][4][112][16] WMMA numerators
constexpr size_t OFF_OPW1 = OFF_NUM  + (size_t)B * NTERM * QPAD * GPAD; // [B][Q][T*H]
constexpr size_t OFF_OPW0 = OFF_OPW1 + (size_t)B * Q * KW_;
constexpr size_t OFF_OPH1 = OFF_OPW0 + (size_t)B * Q * KW_;      // [B][Q][T*W]
constexpr size_t OFF_OPH0 = OFF_OPH1 + (size_t)B * Q * KH_;
constexpr size_t OFF_TPW1 = OFF_OPH0 + (size_t)B * Q * KH_;      // [B][G][T*H]
constexpr size_t OFF_TPW0 = OFF_TPW1 + (size_t)B * G * KW_;
constexpr size_t OFF_TPH1 = OFF_TPW0 + (size_t)B * G * KW_;      // [B][G][T*W]
constexpr size_t OFF_TPH0 = OFF_TPH1 + (size_t)B * G * KH_;
constexpr size_t ZERO_N   = OFF_NUM;                             // region to zero (S + TS + ZPAD)

// ---------------- helpers ----------------
__device__ __forceinline__ float sigm(float x) { return 1.0f / (1.0f + __expf(-x)); }

__device__ __forceinline__ float wred_max(float v) {
#pragma unroll
  for (int o = 16; o; o >>= 1) v = fmaxf(v, __shfl_xor(v, o, 32));
  return v;
}
__device__ __forceinline__ float wred_min(float v) {
#pragma unroll
  for (int o = 16; o; o >>= 1) v = fminf(v, __shfl_xor(v, o, 32));
  return v;
}
__device__ __forceinline__ float wred_sum(float v) {
#pragma unroll
  for (int o = 16; o; o >>= 1) v += __shfl_xor(v, o, 32);
  return v;
}

// ---------------- K0: zero accumulators + zero pad row ----------------
__global__ void k_zero(float* __restrict__ ws) {
  int i = blockIdx.x * 256 + threadIdx.x;
  if (i < (int)ZERO_N) ws[i] = 0.0f;
}

// ---------------- K1: per-query projections (sigmoid(max) trick) ----------------
__global__ __launch_bounds__(256) void k_qproj(const float* __restrict__ pm,
                                               float* __restrict__ ws) {
  const int t = blockIdx.x, q = blockIdx.y, b = blockIdx.z;
  const float* X = pm + (((size_t)(b * Q + q)) * T + t) * HW;
  float* OPW1 = ws + OFF_OPW1 + ((size_t)(b * Q + q)) * KW_ + t * H;
  float* OPW0 = ws + OFF_OPW0 + ((size_t)(b * Q + q)) * KW_ + t * H;
  float* OPH1 = ws + OFF_OPH1 + ((size_t)(b * Q + q)) * KH_ + t * W;
  float* OPH0 = ws + OFF_OPH0 + ((size_t)(b * Q + q)) * KH_ + t * W;

  __shared__ float cmaxS[8][W];
  __shared__ float cminS[8][W];
  const int wave = threadIdx.x >> 5, lane = threadIdx.x & 31;

  float cmax[5], cmin[5];
#pragma unroll
  for (int i = 0; i < 5; ++i) { cmax[i] = -1e30f; cmin[i] = 1e30f; }

  for (int h = wave; h < H; h += 8) {
    float rmax = -1e30f, rmin = 1e30f;
#pragma unroll
    for (int i = 0; i < 5; ++i) {
      float v = X[h * W + i * 32 + lane];
      rmax = fmaxf(rmax, v); rmin = fminf(rmin, v);
      cmax[i] = fmaxf(cmax[i], v); cmin[i] = fminf(cmin[i], v);
    }
    rmax = wred_max(rmax); rmin = wred_min(rmin);
    if (lane == 0) { OPW1[h] = sigm(rmax); OPW0[h] = 1.0f - sigm(rmin); }
  }
#pragma unroll
  for (int i = 0; i < 5; ++i) {
    cmaxS[wave][i * 32 + lane] = cmax[i];
    cminS[wave][i * 32 + lane] = cmin[i];
  }
  __syncthreads();
  for (int w = threadIdx.x; w < W; w += 256) {
    float m = cmaxS[0][w], n = cminS[0][w];
#pragma unroll
    for (int j = 1; j < 8; ++j) { m = fmaxf(m, cmaxS[j][w]); n = fminf(n, cminS[j][w]); }
    OPH1[w] = sigm(m); OPH0[w] = 1.0f - sigm(n);
  }
}

// ---------------- K2: per-target projections + projected sums ----------------
__global__ __launch_bounds__(256) void k_gproj(const float* __restrict__ bm,
                                               float* __restrict__ ws) {
  const int t = blockIdx.x, g = blockIdx.y, b = blockIdx.z;
  const float* X = bm + (((size_t)(b * G + g)) * T + t) * HW;
  float* TPW1 = ws + OFF_TPW1 + ((size_t)(b * G + g)) * KW_ + t * H;
  float* TPW0 = ws + OFF_TPW0 + ((size_t)(b * G + g)) * KW_ + t * H;
  float* TPH1 = ws + OFF_TPH1 + ((size_t)(b * G + g)) * KH_ + t * W;
  float* TPH0 = ws + OFF_TPH0 + ((size_t)(b * G + g)) * KH_ + t * W;
  float* TS   = ws + OFF_TS + (size_t)(b * G + g) * 4;

  __shared__ float cmaxS[8][W];
  __shared__ float cminS[8][W];
  const int wave = threadIdx.x >> 5, lane = threadIdx.x & 31;

  float cmax[5], cmin[5];
#pragma unroll
  for (int i = 0; i < 5; ++i) { cmax[i] = -1e30f; cmin[i] = 1e30f; }

  for (int h = wave; h < H; h += 8) {
    float rmax = -1e30f, rmin = 1e30f;
#pragma unroll
    for (int i = 0; i < 5; ++i) {
      float v = X[h * W + i * 32 + lane];
      rmax = fmaxf(rmax, v); rmin = fminf(rmin, v);
      cmax[i] = fmaxf(cmax[i], v); cmin[i] = fminf(cmin[i], v);
    }
    rmax = wred_max(rmax); rmin = wred_min(rmin);
    if (lane == 0) {
      float p0 = 1.0f - rmin;
      TPW1[h] = rmax; TPW0[h] = p0;
      atomicAdd(&TS[0], rmax);
      atomicAdd(&TS[2], p0);
    }
  }
#pragma unroll
  for (int i = 0; i < 5; ++i) {
    cmaxS[wave][i * 32 + lane] = cmax[i];
    cminS[wave][i * 32 + lane] = cmin[i];
  }
  __syncthreads();
  for (int w = threadIdx.x; w < W; w += 256) {
    float m = cmaxS[0][w], n = cminS[0][w];
#pragma unroll
    for (int j = 1; j < 8; ++j) { m = fmaxf(m, cmaxS[j][w]); n = fminf(n, cminS[j][w]); }
    float p0 = 1.0f - n;
    TPH1[w] = m; TPH0[w] = p0;
    atomicAdd(&TS[1], m);
    atomicAdd(&TS[3], p0);
  }
}

// ---------------- K3: pairwise masked maxes, W axis (terms 0 and 2) ----------------
constexpr int QT = 8;    // queries per block
constexpr int HCH = 4;   // h rows per block
__global__ __launch_bounds__(256) void k_pair_w(const float* __restrict__ pm,
                                                const float* __restrict__ bm,
                                                float* __restrict__ ws) {
  __shared__ float sS[QT][HCH][W];       // 20 KB  sigmoided predictions
  __shared__ float sT[G][HCH][W + 1];    // 30.9 KB (pad row -> conflict-free g stride)
  const int bz = blockIdx.z;
  const int bb = bz / T, tt = bz % T;
  const int h0 = blockIdx.x * HCH;
  const int q0 = blockIdx.y * QT;

  for (int e = threadIdx.x; e < QT * HCH * W; e += 256) {
    int qi = e / (HCH * W), r = e % (HCH * W);
    int hi = r / W, wi = r % W;
    int qg = min(q0 + qi, Q - 1);
    float x = pm[(((size_t)(bb * Q + qg)) * T + tt) * HW + (size_t)(h0 + hi) * W + wi];
    sS[qi][hi][wi] = sigm(x);
  }
  for (int e = threadIdx.x; e < G * HCH * W; e += 256) {
    int gi = e / (HCH * W), r = e % (HCH * W);
    int hi = r / W, wi = r % W;
    sT[gi][hi][wi] = bm[(((size_t)(bb * G + gi)) * T + tt) * HW + (size_t)(h0 + hi) * W + wi];
  }
  __syncthreads();

  const int wave = threadIdx.x >> 5, lane = threadIdx.x & 31;
  const int g = lane & 15, hh = lane >> 4;
  const bool act = (g < G);

  float acc1 = 0.0f, acc0 = 0.0f;
  if (act) {
#pragma unroll
    for (int hp = 0; hp < HCH / 2; ++hp) {
      const int h = hp * 2 + hh;
      float m1 = 0.0f, m0 = 0.0f;
#pragma unroll 4
      for (int w = 0; w < W; ++w) {
        float s = sS[wave][h][w];
        float tv = sT[g][h][w];
        float u = 1.0f - s;
        m1 = fmaxf(m1, s * tv);
        m0 = fmaxf(m0, __builtin_fmaf(-tv, u, u));   // (1-s)*(1-t)
      }
      acc1 += m1; acc0 += m0;
    }
  }
  acc1 += __shfl_xor(acc1, 16, 32);
  acc0 += __shfl_xor(acc0, 16, 32);
  const int qg = q0 + wave;
  if (hh == 0 && act && qg < Q) {
    float* Sb = ws + OFF_S + ((size_t)((bb * Q + qg) * G + g)) * 4;
    atomicAdd(&Sb[0], acc1);
    atomicAdd(&Sb[2], acc0);
  }
}

// ---------------- K4: pairwise masked maxes, H axis (terms 1 and 3) ----------------
constexpr int GQ = 3;     // g per block (4 slices cover G=12)
__global__ __launch_bounds__(256) void k_pair_h(const float* __restrict__ pm,
                                                const float* __restrict__ bm,
                                                float* __restrict__ ws) {
  __shared__ float sT[GQ][H][32];  // 36.9 KB
  const int bz = blockIdx.z;
  const int gq = bz & 3;
  const int bt = bz >> 2;
  const int tt = bt % T, bb = bt / T;
  const int wc = blockIdx.x * 32;
  const int q0 = blockIdx.y * QT;

  for (int e = threadIdx.x; e < GQ * H * 32; e += 256) {
    int gi = e / (H * 32), r = e % (H * 32);
    int hi = r / 32, wi = r & 31;
    sT[gi][hi][wi] =
        bm[(((size_t)(bb * G + gq * GQ + gi)) * T + tt) * HW + (size_t)hi * W + wc + wi];
  }
  __syncthreads();

  const int wave = threadIdx.x >> 5, lane = threadIdx.x & 31;
  const int qg = q0 + wave;
  const bool qok = (qg < Q);
  const int qc = qok ? qg : (Q - 1);
  const float* Xp = pm + (((size_t)(bb * Q + qc)) * T + tt) * HW + wc + lane;

  float m1[GQ] = {0.0f, 0.0f, 0.0f};
  float m0[GQ] = {0.0f, 0.0f, 0.0f};
  for (int h = 0; h < H; ++h) {
    float s = sigm(Xp[(size_t)h * W]);
    float u = 1.0f - s;
#pragma unroll
    for (int gi = 0; gi < GQ; ++gi) {
      float tv = sT[gi][h][lane];
      m1[gi] = fmaxf(m1[gi], s * tv);
      m0[gi] = fmaxf(m0[gi], __builtin_fmaf(-tv, u, u));
    }
  }
#pragma unroll
  for (int gi = 0; gi < GQ; ++gi) {
    float s1 = wred_sum(m1[gi]);
    float s0 = wred_sum(m0[gi]);
    if (lane == 0 && qok) {
      float* Sb = ws + OFF_S + ((size_t)((bb * Q + qg) * G + gq * GQ + gi)) * 4;
      atomicAdd(&Sb[1], s1);
      atomicAdd(&Sb[3], s0);
    }
  }
}

// ---------------- K5: numerator GEMMs via V_WMMA_F32_16X16X4_F32 ----------------
typedef __attribute__((ext_vector_type(2))) float v2f;
typedef __attribute__((ext_vector_type(8))) float v8f;

__global__ __launch_bounds__(32) void k_wmma_num(float* __restrict__ ws) {
  const int qt = blockIdx.x;      // 0..6  (Q padded to 112)
  const int term = blockIdx.y;    // 0..3
  const int b = blockIdx.z;

  const float* Ab; const float* Bb; int K;
  if (term == 0)      { Ab = ws + OFF_OPW1 + (size_t)b * Q * KW_; Bb = ws + OFF_TPW1 + (size_t)b * G * KW_; K = KW_; }
  else if (term == 1) { Ab = ws + OFF_OPH1 + (size_t)b * Q * KH_; Bb = ws + OFF_TPH1 + (size_t)b * G * KH_; K = KH_; }
  else if (term == 2) { Ab = ws + OFF_OPW0 + (size_t)b * Q * KW_; Bb = ws + OFF_TPW0 + (size_t)b * G * KW_; K = KW_; }
  else                { Ab = ws + OFF_OPH0 + (size_t)b * Q * KH_; Bb = ws + OFF_TPH0 + (size_t)b * G * KH_; K = KH_; }

  const int lane = threadIdx.x;
  const int hi = lane >> 4;          // K half: 0 -> K=k0+0..1, 1 -> K=k0+2..3
  const int m = lane & 15;           // A row / B column within tile
  const int qrow = qt * 16 + m;
  const int qc = qrow < Q ? qrow : (Q - 1);   // clamp; garbage rows never read back
  const bool gok = (m < G);

  // Invalid-g lanes read a zeroed scratch row (length KH_ >= K): loads stay
  // unconditional -> clean global_load_b64 + v_wmma pipeline, no EXEC juggling.
  const float* Ap = Ab + (size_t)qc * K + (hi ? 2 : 0);
  const float* Bp = (gok ? (Bb + (size_t)m * K) : (ws + OFF_ZPAD)) + (hi ? 2 : 0);

  v8f c = {};
#pragma unroll 4
  for (int k = 0; k < K; k += 4) {
    float2 av = *(const float2*)(Ap + k);
    float2 bv = *(const float2*)(Bp + k);
    v2f a; a.x = av.x; a.y = av.y;
    v2f bb; bb.x = bv.x; bb.y = bv.y;
    c = __builtin_amdgcn_wmma_f32_16x16x4_f32(false, a, false, bb,
                                              (short)0, c, false, false);
  }

  // D layout: VGPR r holds (M=r, N=lane) for lanes 0..15, (M=r+8, N=lane-16) for 16..31
  float* Np = ws + OFF_NUM + ((size_t)(b * NTERM + term)) * QPAD * GPAD;
#pragma unroll
  for (int r = 0; r < 8; ++r) {
    int row = r + (hi ? 8 : 0);
    Np[(size_t)(qt * 16 + row) * GPAD + m] = c[r];
  }
}

// ---------------- K6: softmax class cost + final assembly ----------------
__global__ __launch_bounds__(32) void k_final(const float* __restrict__ logits,
                                              const int* __restrict__ labels,
                                              const float* __restrict__ ws,
                                              float* __restrict__ out) {
  const int bq = blockIdx.x;
  const int b = bq / Q, q = bq % Q;
  const float* L = logits + (size_t)(b * Q + q) * NC;
  const int lane = threadIdx.x;

  float l1 = (lane < NC) ? L[lane] : -1e30f;
  float l2 = (lane + 32 < NC) ? L[lane + 32] : -1e30f;
  float mx = wred_max(fmaxf(l1, l2));
  float e = ((lane < NC) ? __expf(l1 - mx) : 0.0f) +
            ((lane + 32 < NC) ? __expf(l2 - mx) : 0.0f);
  float denom = wred_sum(e);

  if (lane < G) {
    int lab = labels[b * G + lane];
    float prob = __expf(L[lab] - mx) / denom;
    const float* Sb = ws + OFF_S + ((size_t)((b * Q + q) * G + lane)) * 4;
    const float* TSb = ws + OFF_TS + (size_t)(b * G + lane) * 4;
    float proj = 0.0f;
#pragma unroll
    for (int t = 0; t < 4; ++t) {
      float num = ws[OFF_NUM + ((size_t)(b * NTERM + t) * QPAD + q) * GPAD + lane];
      proj += 1.0f - (2.0f * num + 1.0f) / (Sb[t] + TSb[t] + 1.0f);
    }
    out[(size_t)(b * Q + q) * G + lane] = 2.0f * (-prob) + 5.0f * proj;
  }
}

// ---------------- launcher ----------------
extern "C" void kernel_launch(void* const* d_in, const int* in_sizes, int n_in,
                              void* d_out, int out_size, void* d_ws, size_t ws_size,
                              hipStream_t stream) {
  const float* pred_logits = (const float*)d_in[0];
  const float* pred_masks  = (const float*)d_in[1];
  const float* box_masks   = (const float*)d_in[2];
  const int*   labels      = (const int*)d_in[3];
  float* out = (float*)d_out;
  float* ws  = (float*)d_ws;

  // 0) zero pairwise-sum + tgt-sum accumulators + WMMA zero-pad row
  k_zero<<<dim3(((int)ZERO_N + 255) / 256), dim3(256), 0, stream>>>(ws);
  // 1) per-query projections
  k_qproj<<<dim3(T, Q, B), dim3(256), 0, stream>>>(pred_masks, ws);
  // 2) per-target projections + projected sums
  k_gproj<<<dim3(T, G, B), dim3(256), 0, stream>>>(box_masks, ws);
  // 3) pairwise masked maxes along W (terms 0,2)
  k_pair_w<<<dim3(H / HCH, (Q + QT - 1) / QT, B * T), dim3(256), 0, stream>>>(
      pred_masks, box_masks, ws);
  // 4) pairwise masked maxes along H (terms 1,3)
  k_pair_h<<<dim3(W / 32, (Q + QT - 1) / QT, B * T * (G / GQ)), dim3(256), 0, stream>>>(
      pred_masks, box_masks, ws);
  // 5) WMMA numerator GEMMs
  k_wmma_num<<<dim3(QPAD / 16, NTERM, B), dim3(32), 0, stream>>>(ws);
  // 6) final assembly
  k_final<<<dim3(B * Q), dim3(32), 0, stream>>>(pred_logits, labels, ws, out);
}